// Encoder_40381282517462
// MI455X (gfx1250) — compile-verified
//
#include <hip/hip_runtime.h>
#include <math.h>

typedef __attribute__((ext_vector_type(2))) float v2f;
typedef __attribute__((ext_vector_type(8))) float v8f;

#define EPSV 1e-5f

// ================= WMMA fp32 GEMM: C = alpha*(A·B) + bias, optional ReLU ==========
// TRANSB=true : Bm stored (N x ldb) row-major, b(k,n)=Bm[n*ldb+k]   (weights W[cout][cin])
// TRANSB=false: Bm stored (K x ldb) row-major, b(k,n)=Bm[k*ldb+n]
// Inner loop is guard-free: out-of-range rows/cols are clamped (their products only
// affect D rows/cols that the guarded epilogue never stores), and the K tail relies
// on one guarded step that zero-fills both A and B past K.
template<bool TRANSB>
__global__ __launch_bounds__(128) void wmma_gemm_f32(
    const float* __restrict__ A, const float* __restrict__ Bm,
    const float* __restrict__ bias, float* __restrict__ C,
    int M, int N, int K, int lda, int ldb, int ldc,
    float alpha, int fuseRelu, int tilesTotal)
{
  int ntn  = (N + 15) >> 4;
  int tile = blockIdx.x * 4 + (threadIdx.x >> 5);
  if (tile >= tilesTotal) return;            // wave-uniform exit
  int tm   = (tile / ntn) << 4;
  int tn   = (tile % ntn) << 4;
  int lane = threadIdx.x & 31;
  int half = lane >> 4;                      // K/row half-split per ISA 7.12.2
  int l16  = lane & 15;
  int arow = tm + l16; if (arow > M - 1) arow = M - 1;   // clamp instead of guard
  int bcol = tn + l16; if (bcol > N - 1) bcol = N - 1;
  const float* Ap = A + (long)arow * lda + 2 * half;
  const float* Bp = TRANSB ? (Bm + (long)bcol * ldb + 2 * half)
                           : (Bm + bcol + (long)(2 * half) * ldb);
  int K4 = K & ~3;
  v8f acc = {};
  int kb = 0;
#pragma unroll 4
  for (; kb < K4; kb += 4) {
    v2f a, b;
    a[0] = Ap[kb]; a[1] = Ap[kb + 1];                       // merges to b64
    if (TRANSB) { b[0] = Bp[kb]; b[1] = Bp[kb + 1]; }
    else        { b[0] = Bp[(long)kb * ldb]; b[1] = Bp[(long)kb * ldb + ldb]; }
    acc = __builtin_amdgcn_wmma_f32_16x16x4_f32(false, a, false, b, (short)0, acc, false, false);
  }
  if (kb < K) {                               // single guarded tail step
    v2f a, b;
    a[0] = 0.f; a[1] = 0.f; b[0] = 0.f; b[1] = 0.f;
#pragma unroll
    for (int r = 0; r < 2; ++r) {
      int kk = kb + r + 2 * half;
      if (kk < K) {
        a[r] = Ap[kb + r];
        b[r] = TRANSB ? Bp[kb + r] : Bp[(long)(kb + r) * ldb];
      }
    }
    acc = __builtin_amdgcn_wmma_f32_16x16x4_f32(false, a, false, b, (short)0, acc, false, false);
  }
  int cn = tn + l16;
  if (cn < N) {
    float bb = bias ? bias[cn] : 0.f;
#pragma unroll
    for (int r = 0; r < 8; ++r) {
      int cm = tm + r + 8 * half;
      if (cm < M) {
        float v = acc[r] * alpha + bb;
        if (fuseRelu) v = fmaxf(v, 0.f);
        C[(long)cm * ldc + cn] = v;
      }
    }
  }
}

// ================= Fourier embedding: [xyz, sin(24), cos(24)] = 51 ================
__global__ void fourier_kernel(const float* __restrict__ xyz, float* __restrict__ feats, long total)
{
  long t = (long)blockIdx.x * blockDim.x + threadIdx.x;
  if (t >= total) return;
  const float* p = xyz + t * 3;
  float* o = feats + t * 51;
  o[0] = p[0]; o[1] = p[1]; o[2] = p[2];
  for (int c = 0; c < 3; ++c) {
    float f = 1.f;
    for (int q = 0; q < 8; ++q) {
      float e = p[c] * f;
      o[3  + c * 8 + q] = sinf(e);
      o[27 + c * 8 + q] = cosf(e);
      f *= 2.f;
    }
  }
}

// ================= Farthest point sampling (matches jax scan semantics) ===========
#define FPS_MAXN 4096
__global__ void fps_kernel(const float* __restrict__ xyz, long bstride, int N, int npoint,
                           int* __restrict__ out)
{
  int b = blockIdx.x;
  const float* P = xyz + (long)b * bstride;
  __shared__ float sdist[FPS_MAXN];
  __shared__ float rval[256];
  __shared__ int   ridx[256];
  __shared__ int   scur;
  int tid = threadIdx.x;
  for (int i = tid; i < N; i += blockDim.x) sdist[i] = 1e10f;
  if (tid == 0) { scur = 0; out[(long)b * npoint] = 0; }
  __syncthreads();
  for (int t = 1; t < npoint; ++t) {
    int cur = scur;
    float cx = P[cur * 3], cy = P[cur * 3 + 1], cz = P[cur * 3 + 2];
    float bv = -1.f; int bi = 0;
    for (int i = tid; i < N; i += blockDim.x) {
      float dx = P[i * 3] - cx, dy = P[i * 3 + 1] - cy, dz = P[i * 3 + 2] - cz;
      float d = fminf(sdist[i], dx * dx + dy * dy + dz * dz);
      sdist[i] = d;
      if (d > bv) { bv = d; bi = i; }   // ascending i -> first-max wins ties
    }
    rval[tid] = bv; ridx[tid] = bi;
    __syncthreads();
    for (int s = blockDim.x / 2; s > 0; s >>= 1) {
      if (tid < s) {
        float v2 = rval[tid + s]; int i2 = ridx[tid + s];
        if (v2 > rval[tid] || (v2 == rval[tid] && i2 < ridx[tid])) { rval[tid] = v2; ridx[tid] = i2; }
      }
      __syncthreads();
    }
    if (tid == 0) { scur = ridx[0]; out[(long)b * npoint + t] = ridx[0]; }
    __syncthreads();
  }
}

// ================= gather centroid coords =========================================
__global__ void gather_xyz(const float* __restrict__ P, long bs, const int* __restrict__ idx,
                           int np, float* __restrict__ out, long obs, long total)
{
  long t = (long)blockIdx.x * blockDim.x + threadIdx.x;
  if (t >= total) return;
  int j = (int)(t % np); int b = (int)(t / np);
  int pi = idx[(long)b * np + j];
  const float* p = P + (long)b * bs + (long)pi * 3;
  float* q = out + (long)b * obs + (long)j * 3;
  q[0] = p[0]; q[1] = p[1]; q[2] = p[2];
}

// ================= ball query (first nsample in index order, pad with first) ======
__global__ void query_ball_kernel(const float* __restrict__ xyz, long bsX, int N,
                                  const float* __restrict__ nxyz, long bsN, int np, int ns,
                                  float r2, int* __restrict__ gidx, long total)
{
  long g = (long)blockIdx.x * blockDim.x + threadIdx.x;
  if (g >= total) return;
  int b = (int)(g / np), j = (int)(g % np);
  const float* P = xyz + (long)b * bsX;
  const float* Q = nxyz + (long)b * bsN + (long)j * 3;
  float qx = Q[0], qy = Q[1], qz = Q[2];
  int* row = gidx + g * ns;
  int cnt = 0;
  for (int i = 0; i < N && cnt < ns; ++i) {
    float dx = P[i * 3] - qx, dy = P[i * 3 + 1] - qy, dz = P[i * 3 + 2] - qz;
    float d = dx * dx + dy * dy + dz * dz;
    if (d <= r2) row[cnt++] = i;
  }
  int first = (cnt > 0) ? row[0] : 0;
  for (int s = cnt; s < ns; ++s) row[s] = first;
}

// ================= grouped gather: [xyz-centroid, feats], zero-pad to ldgp ========
__global__ void group_gather(const float* __restrict__ xyz, long bsX,
                             const float* __restrict__ nxyz, long bsN,
                             const float* __restrict__ feats, int fdim, long bsF,
                             const int* __restrict__ gidx, int np, int ns,
                             float* __restrict__ gp, int ldgp, long total)
{
  long t = (long)blockIdx.x * blockDim.x + threadIdx.x;
  if (t >= total) return;
  int j = (int)((t / ns) % np); int b = (int)(t / ((long)ns * np));
  int pi = gidx[t];
  const float* P = xyz + (long)b * bsX + (long)pi * 3;
  const float* Q = nxyz + (long)b * bsN + (long)j * 3;
  const float* F = feats + (long)b * bsF + (long)pi * fdim;
  float* R = gp + t * ldgp;
  R[0] = P[0] - Q[0]; R[1] = P[1] - Q[1]; R[2] = P[2] - Q[2];
  for (int c = 0; c < fdim; ++c) R[3 + c] = F[c];
  for (int c = 3 + fdim; c < ldgp; ++c) R[c] = 0.f;
}

// ================= batchnorm (training stats) + ReLU ==============================
__global__ void bn_stats(const float* __restrict__ X, long M, int C,
                         float* __restrict__ mean, float* __restrict__ var)
{
  int c = blockIdx.x;
  __shared__ float ss[256], ss2[256];
  float s = 0.f, s2 = 0.f;
  for (long r = threadIdx.x; r < M; r += blockDim.x) {
    float v = X[r * C + c]; s += v; s2 += v * v;
  }
  ss[threadIdx.x] = s; ss2[threadIdx.x] = s2;
  __syncthreads();
  for (int k = 128; k > 0; k >>= 1) {
    if (threadIdx.x < k) { ss[threadIdx.x] += ss[threadIdx.x + k]; ss2[threadIdx.x] += ss2[threadIdx.x + k]; }
    __syncthreads();
  }
  if (threadIdx.x == 0) {
    float m = ss[0] / (float)M;
    mean[c] = m;
    var[c] = fmaxf(ss2[0] / (float)M - m * m, 0.f);
  }
}

__global__ void bn_apply(float* __restrict__ X, long M, int C,
                         const float* __restrict__ mean, const float* __restrict__ var,
                         const float* __restrict__ g, const float* __restrict__ be, int relu)
{
  long t = (long)blockIdx.x * blockDim.x + threadIdx.x;
  if (t >= M * C) return;
  int c = (int)(t % C);
  float v = X[t];
  v = (v - mean[c]) * rsqrtf(var[c] + EPSV) * g[c] + be[c];
  if (relu) v = fmaxf(v, 0.f);
  X[t] = v;
}

// ================= max over sample dimension ======================================
__global__ void rowmax_kernel(const float* __restrict__ X, int ldx, int S, int np, int C,
                              float* __restrict__ Y, int ldy, long obs, long total)
{
  long t = (long)blockIdx.x * blockDim.x + threadIdx.x;
  if (t >= total) return;
  int c = (int)(t % C);
  long r = t / C;
  int j = (int)(r % np); int b = (int)(r / np);
  const float* Xp = X + (((long)b * np + j) * (long)S) * ldx + c;
  float m = -3.4e38f;
  for (int s = 0; s < S; ++s) m = fmaxf(m, Xp[(long)s * ldx]);
  Y[(long)b * obs + (long)j * ldy + c] = m;
}

// ================= row softmax ====================================================
__global__ void softmax_rows(float* __restrict__ S, int n)
{
  int row = blockIdx.x, tid = threadIdx.x;
  float* R = S + (long)row * n;
  __shared__ float red[256];
  float m = -3.4e38f;
  for (int i = tid; i < n; i += 256) m = fmaxf(m, R[i]);
  red[tid] = m; __syncthreads();
  for (int k = 128; k > 0; k >>= 1) { if (tid < k) red[tid] = fmaxf(red[tid], red[tid + k]); __syncthreads(); }
  m = red[0]; __syncthreads();
  float s = 0.f;
  for (int i = tid; i < n; i += 256) { float e = expf(R[i] - m); R[i] = e; s += e; }
  red[tid] = s; __syncthreads();
  for (int k = 128; k > 0; k >>= 1) { if (tid < k) red[tid] += red[tid + k]; __syncthreads(); }
  float inv = 1.f / red[0];
  for (int i = tid; i < n; i += 256) R[i] *= inv;
}

// ================= kNN top-16 (matches top_k(-d2) ordering incl. ties) ============
__global__ void knn_topk(const float* __restrict__ xyz, long bs, int N, int* __restrict__ kidx, long total)
{
  long t = (long)blockIdx.x * blockDim.x + threadIdx.x;
  if (t >= total) return;
  int b = (int)(t / N), i = (int)(t % N);
  const float* P = xyz + (long)b * bs;
  float qx = P[i * 3], qy = P[i * 3 + 1], qz = P[i * 3 + 2];
  float bd[16]; int bi[16];
  for (int s = 0; s < 16; ++s) { bd[s] = 3.4e38f; bi[s] = 0; }
  for (int j = 0; j < N; ++j) {
    float dx = P[j * 3] - qx, dy = P[j * 3 + 1] - qy, dz = P[j * 3 + 2] - qz;
    float d = dx * dx + dy * dy + dz * dz;
    if (d < bd[15]) {
      int p = 15;
      while (p > 0 && d < bd[p - 1]) { bd[p] = bd[p - 1]; bi[p] = bi[p - 1]; --p; }
      bd[p] = d; bi[p] = j;
    }
  }
  for (int s = 0; s < 16; ++s) kidx[t * 16 + s] = bi[s];
}

// ================= misc elementwise ===============================================
__global__ void gather_rows(const float* __restrict__ feats, int fdim, const int* __restrict__ idx,
                            float* __restrict__ out, long total /*rows*fdim*/)
{
  long t = (long)blockIdx.x * blockDim.x + threadIdx.x;
  if (t >= total) return;
  long r = t / fdim; int c = (int)(t % fdim);
  out[t] = feats[(long)idx[r] * fdim + c];
}

__global__ void add2(const float* __restrict__ a, const float* __restrict__ b,
                     float* __restrict__ o, long n)
{
  long t = (long)blockIdx.x * blockDim.x + threadIdx.x;
  if (t < n) o[t] = a[t] + b[t];
}

__global__ void concat2(const float* __restrict__ A, const float* __restrict__ B,
                        float* __restrict__ o, int Ca, int Cb, long total)
{
  long t = (long)blockIdx.x * blockDim.x + threadIdx.x;
  if (t >= total) return;
  int Ct = Ca + Cb;
  long r = t / Ct; int c = (int)(t % Ct);
  o[t] = (c < Ca) ? A[r * Ca + c] : B[r * Cb + (c - Ca)];
}

__global__ void layernorm_rows(const float* __restrict__ X, float* __restrict__ Y,
                               const float* __restrict__ g, const float* __restrict__ be, int C)
{
  int row = blockIdx.x, tid = threadIdx.x;   // blockDim == C == 128
  __shared__ float red[128];
  float v = X[(long)row * C + tid];
  red[tid] = v; __syncthreads();
  for (int k = 64; k > 0; k >>= 1) { if (tid < k) red[tid] += red[tid + k]; __syncthreads(); }
  float mu = red[0] / (float)C; __syncthreads();
  float d = v - mu;
  red[tid] = d * d; __syncthreads();
  for (int k = 64; k > 0; k >>= 1) { if (tid < k) red[tid] += red[tid + k]; __syncthreads(); }
  float var = red[0] / (float)C;
  Y[(long)row * C + tid] = d * rsqrtf(var + EPSV) * g[tid] + be[tid];
}

__global__ void sa3_gather(const float* __restrict__ l2x, const float* __restrict__ l2p,
                           float* __restrict__ out, int ld)
{
  int t = blockIdx.x * blockDim.x + threadIdx.x;
  if (t >= 512) return;
  float* R = out + (long)t * ld;
  const float* x = l2x + (long)t * 3;
  const float* p = l2p + (long)t * 256;
  R[0] = x[0]; R[1] = x[1]; R[2] = x[2];
  for (int c = 0; c < 256; ++c) R[3 + c] = p[c];
  for (int c = 259; c < ld; ++c) R[c] = 0.f;
}

// ================= host-side helpers ==============================================
static inline unsigned blks(long n) { return (unsigned)((n + 255) / 256); }

static inline void gemmT(hipStream_t s, const float* A, const float* W, const float* bias, float* C,
                         int M, int N, int K, int lda, int ldb, int ldc, float alpha = 1.f, int relu = 0)
{
  int tiles = ((M + 15) / 16) * ((N + 15) / 16);
  wmma_gemm_f32<true><<<dim3((unsigned)((tiles + 3) / 4)), dim3(128), 0, s>>>(
      A, W, bias, C, M, N, K, lda, ldb, ldc, alpha, relu, tiles);
}
static inline void gemmN(hipStream_t s, const float* A, const float* Bm, const float* bias, float* C,
                         int M, int N, int K, int lda, int ldb, int ldc, float alpha = 1.f, int relu = 0)
{
  int tiles = ((M + 15) / 16) * ((N + 15) / 16);
  wmma_gemm_f32<false><<<dim3((unsigned)((tiles + 3) / 4)), dim3(128), 0, s>>>(
      A, Bm, bias, C, M, N, K, lda, ldb, ldc, alpha, relu, tiles);
}
static inline void bnrelu(hipStream_t s, float* X, long M, int C, float* mean, float* var,
                          const float* g, const float* be)
{
  bn_stats<<<dim3((unsigned)C), dim3(256), 0, s>>>(X, M, C, mean, var);
  bn_apply<<<dim3(blks(M * (long)C)), dim3(256), 0, s>>>(X, M, C, mean, var, g, be, 1);
}

struct SAW { const float *W[3], *b[3], *g[3], *be[3]; };

extern "C" void kernel_launch(void* const* d_in, const int* in_sizes, int n_in,
                              void* d_out, int out_size, void* d_ws, size_t ws_size,
                              hipStream_t stream)
{
  (void)in_sizes; (void)n_in; (void)out_size; (void)ws_size;
  // ---------------- unpack inputs (setup_inputs dict order, depth-first) ----------
  int ci = 0;
  const float* xyz = (const float*)d_in[ci++];
  auto rdSA = [&](SAW& s) {
    for (int l = 0; l < 3; ++l) { s.W[l] = (const float*)d_in[ci++]; s.b[l] = (const float*)d_in[ci++]; }
    for (int l = 0; l < 3; ++l) { s.g[l] = (const float*)d_in[ci++]; s.be[l] = (const float*)d_in[ci++]; }
  };
  SAW sa1[3]; rdSA(sa1[0]); rdSA(sa1[1]); rdSA(sa1[2]);
  const float *Wqkv = (const float*)d_in[ci++], *bqkv = (const float*)d_in[ci++];
  const float *Wo   = (const float*)d_in[ci++], *bo   = (const float*)d_in[ci++];
  const float *Wm   = (const float*)d_in[ci++], *bm   = (const float*)d_in[ci++];
  const float *Wk1  = (const float*)d_in[ci++], *bk1  = (const float*)d_in[ci++];
  const float *Wk2  = (const float*)d_in[ci++], *bk2  = (const float*)d_in[ci++];
  const float *Wc   = (const float*)d_in[ci++], *bc   = (const float*)d_in[ci++];
  const float *gln  = (const float*)d_in[ci++], *bln  = (const float*)d_in[ci++];
  const float *Wf   = (const float*)d_in[ci++], *bf   = (const float*)d_in[ci++];
  const float *gfu  = (const float*)d_in[ci++], *befu = (const float*)d_in[ci++];
  SAW sa2; rdSA(sa2);
  SAW sa3; rdSA(sa3);
  const float *Wmu = (const float*)d_in[ci++], *bmu = (const float*)d_in[ci++];
  const float *Wlv = (const float*)d_in[ci++], *blv = (const float*)d_in[ci++];
  float* out = (float*)d_out;

  // ---------------- dimensions ---------------------------------------------------
  const long B = 4, NTOT = 12000, NG = 4000, FD = 51;
  const long NP1 = 1024, NS1 = 32, L1N = 3072;
  const long R1 = B * NP1 * NS1;                 // 131072 SA1 rows

  // ---------------- workspace arena (floats then ints) ---------------------------
  float* F = (float*)d_ws;
  size_t o = 0;
  auto al = [&](size_t n) { size_t r = o; o += n; return r; };
  size_t f_feats = al(B * NTOT * FD);
  size_t f_big1  = al(R1 * 64);
  size_t f_big2  = al(R1 * 64);
  size_t f_big3  = al(R1 * 128);
  size_t f_l1x   = al(B * L1N * 3);
  size_t f_l1p   = al(B * L1N * 128);
  size_t f_a     = al(B * L1N * 128);
  size_t f_aO    = al(B * L1N * 128);
  size_t f_gF    = al(B * L1N * 128);
  size_t f_lM    = al(B * L1N * 128);
  size_t f_lF    = al(B * L1N * 128);
  size_t f_cat   = al(B * L1N * 256);
  size_t f_attn  = al(B * L1N * 128);
  size_t f_fused = al(B * L1N * 128);
  size_t f_l2x   = al(B * 128 * 3);
  size_t f_l2p   = al(B * 128 * 256);
  size_t f_gf    = al(B * 1024);
  size_t f_mean  = al(1024);
  size_t f_var   = al(1024);
  int* I = (int*)(F + o);
  size_t io = 0;
  auto ial = [&](size_t n) { size_t r = io; io += n; return r; };
  size_t i_fidx  = ial(B * NP1);
  size_t i_gidx1 = ial(B * NP1 * NS1);
  size_t i_kidx  = ial(B * L1N * 16);
  size_t i_fidx2 = ial(B * 128);
  size_t i_gidx2 = ial(B * 128 * 64);

  float *feats = F + f_feats, *big1 = F + f_big1, *big2 = F + f_big2, *big3 = F + f_big3;
  float *l1x = F + f_l1x, *l1p = F + f_l1p, *aBuf = F + f_a, *aO = F + f_aO;
  float *gFb = F + f_gF, *lM = F + f_lM, *lFb = F + f_lF, *cat = F + f_cat;
  float *attnOut = F + f_attn, *fused = F + f_fused, *l2x = F + f_l2x, *l2p = F + f_l2p;
  float *gf = F + f_gf, *meanB = F + f_mean, *varB = F + f_var;
  int *fidx = I + i_fidx, *gidx1 = I + i_gidx1, *kidx = I + i_kidx;
  int *fidx2 = I + i_fidx2, *gidx2 = I + i_gidx2;

  // ---------------- 1. Fourier embed ---------------------------------------------
  fourier_kernel<<<dim3(blks(B * NTOT)), dim3(256), 0, stream>>>(xyz, feats, B * NTOT);

  // ---------------- 2. SA1 per point group ---------------------------------------
  for (int g = 0; g < 3; ++g) {
    const float* Pg = xyz + (long)g * NG * 3;                   // batch stride 36000
    const float* Fg = feats + (long)g * NG * FD;                // batch stride 12000*51
    fps_kernel<<<dim3(4), dim3(256), 0, stream>>>(Pg, NTOT * 3, (int)NG, (int)NP1, fidx);
    gather_xyz<<<dim3(blks(B * NP1)), dim3(256), 0, stream>>>(
        Pg, NTOT * 3, fidx, (int)NP1, l1x + (long)g * NP1 * 3, L1N * 3, B * NP1);
    query_ball_kernel<<<dim3(blks(B * NP1)), dim3(256), 0, stream>>>(
        Pg, NTOT * 3, (int)NG, l1x + (long)g * NP1 * 3, L1N * 3, (int)NP1, (int)NS1,
        0.2f * 0.2f, gidx1, B * NP1);
    group_gather<<<dim3(blks(R1)), dim3(256), 0, stream>>>(
        Pg, NTOT * 3, l1x + (long)g * NP1 * 3, L1N * 3, Fg, (int)FD, NTOT * FD,
        gidx1, (int)NP1, (int)NS1, big1, 64, R1);
    gemmT(stream, big1, sa1[g].W[0], sa1[g].b[0], big2, (int)R1, 64, 54, 64, 54, 64);
    bnrelu(stream, big2, R1, 64, meanB, varB, sa1[g].g[0], sa1[g].be[0]);
    gemmT(stream, big2, sa1[g].W[1], sa1[g].b[1], big1, (int)R1, 64, 64, 64, 64, 64);
    bnrelu(stream, big1, R1, 64, meanB, varB, sa1[g].g[1], sa1[g].be[1]);
    gemmT(stream, big1, sa1[g].W[2], sa1[g].b[2], big3, (int)R1, 128, 64, 64, 64, 128);
    bnrelu(stream, big3, R1, 128, meanB, varB, sa1[g].g[2], sa1[g].be[2]);
    rowmax_kernel<<<dim3(blks(B * NP1 * 128)), dim3(256), 0, stream>>>(
        big3, 128, (int)NS1, (int)NP1, 128, l1p + (long)g * NP1 * 128, 128,
        L1N * 128, B * NP1 * 128);
  }

  // ---------------- 3. MHSA ------------------------------------------------------
  float* qkv = big1;                                            // 12288 x 384
  gemmT(stream, l1p, Wqkv, bqkv, qkv, (int)(B * L1N), 384, 128, 128, 128, 384);
  float* Sb = big3;                                             // 3072 x 3072 reused per (b,h)
  const float scal = 0.1767766952966369f;                       // 1/sqrt(32)
  for (int b = 0; b < 4; ++b)
    for (int h = 0; h < 4; ++h) {
      const float* Qp = qkv + (long)b * L1N * 384 + h * 32;
      const float* Kp = Qp + 128;
      const float* Vp = Qp + 256;
      gemmT(stream, Qp, Kp, nullptr, Sb, (int)L1N, (int)L1N, 32, 384, 384, (int)L1N, scal);
      softmax_rows<<<dim3((unsigned)L1N), dim3(256), 0, stream>>>(Sb, (int)L1N);
      gemmN(stream, Sb, Vp, nullptr, aBuf + (long)b * L1N * 128 + h * 32,
            (int)L1N, 32, (int)L1N, (int)L1N, 384, 128);
    }
  gemmT(stream, aBuf, Wo, bo, aO, (int)(B * L1N), 128, 128, 128, 128, 128);
  gemmT(stream, aO, Wm, bm, gFb, (int)(B * L1N), 128, 128, 128, 128, 128);

  // ---------------- 4. kNN local branch ------------------------------------------
  knn_topk<<<dim3(blks(B * L1N)), dim3(256), 0, stream>>>(l1x, L1N * 3, (int)L1N, kidx, B * L1N);
  for (int b = 0; b < 4; ++b) {
    long rows = L1N * 16;
    gather_rows<<<dim3(blks(rows * 128)), dim3(256), 0, stream>>>(
        l1p + (long)b * L1N * 128, 128, kidx + (long)b * L1N * 16, big1, rows * 128);
    gemmT(stream, big1, Wk1, bk1, big2, (int)rows, 128, 128, 128, 128, 128, 1.f, 1);
    rowmax_kernel<<<dim3(blks(L1N * 128)), dim3(256), 0, stream>>>(
        big2, 128, 16, (int)L1N, 128, lM + (long)b * L1N * 128, 128, 0, L1N * 128);
  }
  gemmT(stream, lM, Wk2, bk2, lFb, (int)(B * L1N), 128, 128, 128, 128, 128);

  // ---------------- 5. concat + fuse + residual + LN -----------------------------
  concat2<<<dim3(blks(B * L1N * 256)), dim3(256), 0, stream>>>(gFb, lFb, cat, 128, 128, B * L1N * 256);
  gemmT(stream, cat, Wc, bc, gFb, (int)(B * L1N), 128, 256, 256, 256, 128, 1.f, 1);
  add2<<<dim3(blks(B * L1N * 128)), dim3(256), 0, stream>>>(gFb, l1p, aO, B * L1N * 128);
  layernorm_rows<<<dim3((unsigned)(B * L1N)), dim3(128), 0, stream>>>(aO, attnOut, gln, bln, 128);

  // ---------------- 6. fusion ----------------------------------------------------
  add2<<<dim3(blks(B * L1N * 128)), dim3(256), 0, stream>>>(l1p, attnOut, aO, B * L1N * 128);
  gemmT(stream, aO, Wf, bf, fused, (int)(B * L1N), 128, 128, 128, 128, 128);
  bnrelu(stream, fused, B * L1N, 128, meanB, varB, gfu, befu);

  // ---------------- 7. SA2 -------------------------------------------------------
  const long NP2 = 128, NS2 = 64, R2 = B * NP2 * NS2;           // 32768 rows
  fps_kernel<<<dim3(4), dim3(256), 0, stream>>>(l1x, L1N * 3, (int)L1N, (int)NP2, fidx2);
  gather_xyz<<<dim3(blks(B * NP2)), dim3(256), 0, stream>>>(
      l1x, L1N * 3, fidx2, (int)NP2, l2x, NP2 * 3, B * NP2);
  query_ball_kernel<<<dim3(blks(B * NP2)), dim3(256), 0, stream>>>(
      l1x, L1N * 3, (int)L1N, l2x, NP2 * 3, (int)NP2, (int)NS2, 0.4f * 0.4f, gidx2, B * NP2);
  group_gather<<<dim3(blks(R2)), dim3(256), 0, stream>>>(
      l1x, L1N * 3, l2x, NP2 * 3, fused, 128, L1N * 128, gidx2, (int)NP2, (int)NS2, big1, 144, R2);
  gemmT(stream, big1, sa2.W[0], sa2.b[0], big2, (int)R2, 128, 131, 144, 131, 128);
  bnrelu(stream, big2, R2, 128, meanB, varB, sa2.g[0], sa2.be[0]);
  gemmT(stream, big2, sa2.W[1], sa2.b[1], big3, (int)R2, 128, 128, 128, 128, 128);
  bnrelu(stream, big3, R2, 128, meanB, varB, sa2.g[1], sa2.be[1]);
  gemmT(stream, big3, sa2.W[2], sa2.b[2], big1, (int)R2, 256, 128, 128, 128, 256);
  bnrelu(stream, big1, R2, 256, meanB, varB, sa2.g[2], sa2.be[2]);
  rowmax_kernel<<<dim3(blks(B * NP2 * 256)), dim3(256), 0, stream>>>(
      big1, 256, (int)NS2, (int)NP2, 256, l2p, 256, NP2 * 256, B * NP2 * 256);

  // ---------------- 8. SA3 (group-all) -------------------------------------------
  sa3_gather<<<dim3(2), dim3(256), 0, stream>>>(l2x, l2p, big2, 272);
  gemmT(stream, big2, sa3.W[0], sa3.b[0], big3, 512, 256, 259, 272, 259, 256);
  bnrelu(stream, big3, 512, 256, meanB, varB, sa3.g[0], sa3.be[0]);
  gemmT(stream, big3, sa3.W[1], sa3.b[1], big1, 512, 512, 256, 256, 256, 512);
  bnrelu(stream, big1, 512, 512, meanB, varB, sa3.g[1], sa3.be[1]);
  gemmT(stream, big1, sa3.W[2], sa3.b[2], big2, 512, 1024, 512, 512, 512, 1024);
  bnrelu(stream, big2, 512, 1024, meanB, varB, sa3.g[2], sa3.be[2]);
  rowmax_kernel<<<dim3(blks(B * 1024)), dim3(256), 0, stream>>>(
      big2, 1024, 128, 1, 1024, gf, 1024, 1024, B * 1024);

  // ---------------- 9. FC heads --> d_out ----------------------------------------
  gemmT(stream, gf, Wmu, bmu, out,       4, 128, 1024, 1024, 1024, 128);
  gemmT(stream, gf, Wlv, blv, out + 512, 4, 128, 1024, 1024, 1024, 128);
}